// AttnDecoderRNN_33578054320464
// MI455X (gfx1250) — compile-verified
//
#include <hip/hip_runtime.h>
#include <math.h>

// Problem dims (match reference)
#define HD 1024     // hidden
#define VD 50257    // vocab
#define LD 512      // max length

typedef float v2f __attribute__((ext_vector_type(2)));
typedef float v8f __attribute__((ext_vector_type(8)));

__device__ __forceinline__ float wave_reduce_sum(float s) {
#pragma unroll
    for (int off = 16; off > 0; off >>= 1) s += __shfl_xor(s, off, 32);
    return s;
}

// K1: attn_logits[l] = dot(concat(e, h0), attn_W[l,:]) + attn_b[l]
// one wave per row l; 64 blocks x 256 threads = 512 waves
__global__ void k1_attn_logits(const int* __restrict__ x,
                               const float* __restrict__ hidden,
                               const float* __restrict__ emb,
                               const float* __restrict__ attn_W,
                               const float* __restrict__ attn_b,
                               float* __restrict__ logits) {
    const int wave = threadIdx.x >> 5, lane = threadIdx.x & 31;
    const int l = blockIdx.x * 8 + wave;
    if (l >= LD) return;
    const float* erow = emb + (size_t)x[0] * HD;
    const float* wrow = attn_W + (size_t)l * (2 * HD);
    float s = 0.f;
    for (int k = lane; k < HD; k += 32)
        s += erow[k] * wrow[k] + hidden[k] * wrow[HD + k];
    s = wave_reduce_sum(s);
    if (lane == 0) logits[l] = s + attn_b[l];
}

// K2: softmax over 512 logits; single block of 512 threads
__global__ void k2_softmax(const float* __restrict__ logits,
                           float* __restrict__ wts) {
    __shared__ float red[LD];
    const int t = threadIdx.x;
    const float v = logits[t];
    red[t] = v;
    __syncthreads();
    for (int s = LD / 2; s > 0; s >>= 1) {
        if (t < s) red[t] = fmaxf(red[t], red[t + s]);
        __syncthreads();
    }
    const float m = red[0];
    __syncthreads();
    const float e = expf(v - m);
    red[t] = e;
    __syncthreads();
    for (int s = LD / 2; s > 0; s >>= 1) {
        if (t < s) red[t] += red[t + s];
        __syncthreads();
    }
    wts[t] = e / red[0];
}

// K3: ctx[h] = sum_l wts[l] * enc[l,h];  4 blocks x 256, thread per column
__global__ void k3_context(const float* __restrict__ wts,
                           const float* __restrict__ enc,
                           float* __restrict__ ctx) {
    __shared__ float sw[LD];
    for (int i = threadIdx.x; i < LD; i += blockDim.x) sw[i] = wts[i];
    __syncthreads();
    const int h = blockIdx.x * 256 + threadIdx.x;
    if (h >= HD) return;
    float s = 0.f;
    for (int l = 0; l < LD; ++l) s += sw[l] * enc[(size_t)l * HD + h];
    ctx[h] = s;
}

// K4: g_in[j] = relu(dot(concat(e, ctx), comb_W[j,:]) + comb_b[j])
// one wave per row j; 128 blocks x 256 threads = 1024 waves
__global__ void k4_combine(const int* __restrict__ x,
                           const float* __restrict__ emb,
                           const float* __restrict__ ctx,
                           const float* __restrict__ comb_W,
                           const float* __restrict__ comb_b,
                           float* __restrict__ gin) {
    const int wave = threadIdx.x >> 5, lane = threadIdx.x & 31;
    const int j = blockIdx.x * 8 + wave;
    if (j >= HD) return;
    const float* erow = emb + (size_t)x[0] * HD;
    const float* wrow = comb_W + (size_t)j * (2 * HD);
    float s = 0.f;
    for (int k = lane; k < HD; k += 32)
        s += erow[k] * wrow[k] + ctx[k] * wrow[HD + k];
    s = wave_reduce_sum(s);
    if (lane == 0) gin[j] = fmaxf(s + comb_b[j], 0.f);
}

// K5: GRU cell, one wave per hidden index j (6 dots of length H)
__global__ void k5_gru(const float* __restrict__ gin,
                       const float* __restrict__ hidden,
                       const float* __restrict__ W_ih,
                       const float* __restrict__ W_hh,
                       const float* __restrict__ b_ih,
                       const float* __restrict__ b_hh,
                       float* __restrict__ h1,
                       float* __restrict__ out_h) {
    const int wave = threadIdx.x >> 5, lane = threadIdx.x & 31;
    const int j = blockIdx.x * 8 + wave;
    if (j >= HD) return;
    const float* wi0 = W_ih + (size_t)j * HD;
    const float* wi1 = W_ih + (size_t)(HD + j) * HD;
    const float* wi2 = W_ih + (size_t)(2 * HD + j) * HD;
    const float* wh0 = W_hh + (size_t)j * HD;
    const float* wh1 = W_hh + (size_t)(HD + j) * HD;
    const float* wh2 = W_hh + (size_t)(2 * HD + j) * HD;
    float a0 = 0.f, a1 = 0.f, a2 = 0.f, a3 = 0.f, a4 = 0.f, a5 = 0.f;
    for (int k = lane; k < HD; k += 32) {
        const float g = gin[k], h = hidden[k];
        a0 += g * wi0[k]; a1 += g * wi1[k]; a2 += g * wi2[k];
        a3 += h * wh0[k]; a4 += h * wh1[k]; a5 += h * wh2[k];
    }
    a0 = wave_reduce_sum(a0); a1 = wave_reduce_sum(a1); a2 = wave_reduce_sum(a2);
    a3 = wave_reduce_sum(a3); a4 = wave_reduce_sum(a4); a5 = wave_reduce_sum(a5);
    if (lane == 0) {
        const float ir = a0 + b_ih[j],           hr = a3 + b_hh[j];
        const float iz = a1 + b_ih[HD + j],      hz = a4 + b_hh[HD + j];
        const float in_ = a2 + b_ih[2 * HD + j], hn = a5 + b_hh[2 * HD + j];
        const float r = 1.f / (1.f + expf(-(ir + hr)));
        const float z = 1.f / (1.f + expf(-(iz + hz)));
        const float n = tanhf(in_ + r * hn);
        const float hv = (1.f - z) * n + z * hidden[j];
        h1[j] = hv;
        out_h[j] = hv;
    }
}

// K6: logits[v] = dot(h1, out_W[v,:]) + out_b[v], via V_WMMA_F32_16X16X4_F32.
// Each wave owns 16 vocab rows. A = out_W tile (16x4), B = broadcast h1 chunk
// (all 16 columns equal), so D's column 0 holds the 16 row-dots.
// EXEC must stay all-ones for WMMA: clamp loads, guard only the final store.
// out_W (206 MB, single-use, > 192 MB L2) is loaded non-temporal so it does
// not evict the ~38 MB of reusable weights that fit in L2 across replays.
__global__ void k6_out_proj_wmma(const float* __restrict__ W,
                                 const float* __restrict__ bias,
                                 const float* __restrict__ h1,
                                 float* __restrict__ out) {
    __shared__ __align__(16) float sh[HD];
    for (int i = threadIdx.x; i < HD; i += blockDim.x) sh[i] = h1[i];
    __syncthreads();

    const int wave = threadIdx.x >> 5;
    const int lane = threadIdx.x & 31;
    const int row_base = (blockIdx.x * 8 + wave) * 16;
    const int m = lane & 15;
    const int koff = (lane >> 4) << 1;           // lanes 0-15 -> K{0,1}, 16-31 -> K{2,3}
    int row = row_base + m;
    if (row >= VD) row = VD - 1;                 // clamp so EXEC stays full
    const float* wrow = W + (size_t)row * HD;

    v8f acc = {};
#pragma unroll 4
    for (int c = 0; c < HD; c += 4) {
        // A tile: streaming, single-use -> non-temporal (TH_NT)
        v2f a = __builtin_nontemporal_load((const v2f*)(wrow + c + koff));
        v2f b = *(const v2f*)(sh + c + koff);    // B 4x16 broadcast of h1 chunk
        acc = __builtin_amdgcn_wmma_f32_16x16x4_f32(
            false, a, false, b, (short)0, acc, false, false);
    }
    // D: VGPR j = row j (lanes 0-15) / row j+8 (lanes 16-31); columns identical.
    if (lane == 0 || lane == 16) {
        const int rbase = row_base + ((lane >> 4) << 3);
#pragma unroll
        for (int j = 0; j < 8; ++j) {
            const int v = rbase + j;
            if (v < VD) {
                const float val = acc[j] + bias[v];
                __builtin_nontemporal_store(val, out + v);  // write-once output
            }
        }
    }
}

extern "C" void kernel_launch(void* const* d_in, const int* in_sizes, int n_in,
                              void* d_out, int out_size, void* d_ws, size_t ws_size,
                              hipStream_t stream) {
    const int*   x       = (const int*)  d_in[0];
    const float* hidden  = (const float*)d_in[1];   // [1,1,H]
    const float* enc     = (const float*)d_in[2];   // [L,H]
    const float* emb     = (const float*)d_in[3];   // [V,H]
    const float* attn_W  = (const float*)d_in[4];   // [L,2H]
    const float* attn_b  = (const float*)d_in[5];   // [L]
    const float* comb_W  = (const float*)d_in[6];   // [H,2H]
    const float* comb_b  = (const float*)d_in[7];   // [H]
    const float* W_ih    = (const float*)d_in[8];   // [3H,H]
    const float* W_hh    = (const float*)d_in[9];   // [3H,H]
    const float* b_ih    = (const float*)d_in[10];  // [3H]
    const float* b_hh    = (const float*)d_in[11];  // [3H]
    const float* out_W   = (const float*)d_in[12];  // [V,H]
    const float* out_b   = (const float*)d_in[13];  // [V]

    float* out_logits = (float*)d_out;              // [V]
    float* out_h      = (float*)d_out + VD;         // [H]

    float* ws      = (float*)d_ws;
    float* logits  = ws;                            // 512
    float* wts     = ws + LD;                       // 512
    float* ctx     = ws + 2 * LD;                   // 1024
    float* gin     = ctx + HD;                      // 1024
    float* h1      = gin + HD;                      // 1024

    k1_attn_logits<<<LD / 8, 256, 0, stream>>>(x, hidden, emb, attn_W, attn_b, logits);
    k2_softmax<<<1, LD, 0, stream>>>(logits, wts);
    k3_context<<<HD / 256, 256, 0, stream>>>(wts, enc, ctx);
    k4_combine<<<HD / 8, 256, 0, stream>>>(x, emb, ctx, comb_W, comb_b, gin);
    k5_gru<<<HD / 8, 256, 0, stream>>>(gin, hidden, W_ih, W_hh, b_ih, b_hh, h1, out_h);
    const int nb6 = (VD + 127) / 128;               // 128 rows per block (8 waves x 16)
    k6_out_proj_wmma<<<nb6, 256, 0, stream>>>(out_W, out_b, h1, out_logits);
}